// Bernprop2_14654428414711
// MI455X (gfx1250) — compile-verified
//
#include <hip/hip_runtime.h>
#include <cstddef>

typedef __attribute__((ext_vector_type(2))) float v2f;
typedef __attribute__((ext_vector_type(8))) float v8f;

// ---------------------------------------------------------------------------
// elementwise helpers
// ---------------------------------------------------------------------------
__global__ void zero_kernel(float* __restrict__ p, size_t n) {
    size_t i = (size_t)blockIdx.x * blockDim.x + threadIdx.x;
    if (i < n) p[i] = 0.0f;
}

__global__ void sub_kernel(const float* __restrict__ a, const float* __restrict__ b,
                           float* __restrict__ o, size_t n) {
    size_t i = (size_t)blockIdx.x * blockDim.x + threadIdx.x;
    if (i < n) o[i] = a[i] - b[i];
}

// deg[row[e]] += ew[e]
__global__ void deg_kernel(const int* __restrict__ row, const float* __restrict__ ew,
                           float* __restrict__ deg, int E) {
    int e = blockIdx.x * blockDim.x + threadIdx.x;
    if (e < E) unsafeAtomicAdd(&deg[row[e]], ew[e]);
}

// deg -> dis = deg>0 ? 1/sqrt(deg) : 0   (in place)
__global__ void invsqrt_kernel(float* __restrict__ deg, int N) {
    int i = blockIdx.x * blockDim.x + threadIdx.x;
    if (i < N) {
        float d = deg[i];
        deg[i] = (d > 0.0f) ? (1.0f / sqrtf(d)) : 0.0f;
    }
}

// ---------------------------------------------------------------------------
// SpMM: one wave (32 lanes) per edge, D = 64, each lane owns 2 features.
// dst[row] += (dis[row]*ew*dis[col]) * src[col]
// ---------------------------------------------------------------------------
__global__ void spmm_norm_kernel(const int* __restrict__ row, const int* __restrict__ col,
                                 const float* __restrict__ ew, const float* __restrict__ dis,
                                 const float* __restrict__ src, float* __restrict__ dst, int E) {
    int wid  = (int)((blockIdx.x * (size_t)blockDim.x + threadIdx.x) >> 5);
    if (wid >= E) return;
    int lane = threadIdx.x & 31;
    int r = row[wid];
    int c = col[wid];
    float w = dis[r] * ew[wid] * dis[c];
    v2f v = *(const v2f*)(src + (size_t)c * 64 + lane * 2);
    float* dp = dst + (size_t)r * 64 + lane * 2;
    unsafeAtomicAdd(dp,     w * v.x);
    unsafeAtomicAdd(dp + 1, w * v.y);
}

// dst[row] += w * src[gather ? gather[col] : col]
__global__ void spmm_nb_kernel(const int* __restrict__ row, const int* __restrict__ col,
                               const float* __restrict__ ew, const int* __restrict__ gather,
                               const float* __restrict__ src, float* __restrict__ dst, int E) {
    int wid  = (int)((blockIdx.x * (size_t)blockDim.x + threadIdx.x) >> 5);
    if (wid >= E) return;
    int lane = threadIdx.x & 31;
    int r = row[wid];
    int c = col[wid];
    if (gather) c = gather[c];
    float w = ew[wid];
    v2f v = *(const v2f*)(src + (size_t)c * 64 + lane * 2);
    float* dp = dst + (size_t)r * 64 + lane * 2;
    unsafeAtomicAdd(dp,     w * v.x);
    unsafeAtomicAdd(dp + 1, w * v.y);
}

// ---------------------------------------------------------------------------
// WMMA combine: out = ca*X + cb*LX + cc*S2 on 16x16 f32 tiles via
// V_WMMA_F32_16X16X4_F32.  coef*V(16x16) = sum_k V[:,4k:4k+4] x (coef*P_k),
// P_k[j][n] = (n == 4k+j).  One wave per tile; wave-uniform guard keeps EXEC
// all-ones (WMMA requirement).  Requires N%16==0 and D==64.
// ---------------------------------------------------------------------------
__global__ void combine_wmma_kernel(const float* __restrict__ X,
                                    const float* __restrict__ LX,
                                    const float* __restrict__ S2,
                                    const float* __restrict__ temp,
                                    float* __restrict__ out, int nTiles) {
    int wid = (int)((blockIdx.x * (size_t)blockDim.x + threadIdx.x) >> 5);
    if (wid >= nTiles) return;          // uniform per wave
    int lane = threadIdx.x & 31;
    int tm = wid >> 2;                  // row tile (D=64 -> 4 col tiles)
    int tc = wid & 3;
    int R  = tm * 16;
    int Cb = tc * 16;
    int l16 = lane & 15;
    int hi  = lane >> 4;

    // T = relu(temp); out = T0*x + (T1-T0)*Lx + c2*(Lx - S2)
    float T0 = fmaxf(temp[0], 0.0f);
    float T1 = fmaxf(temp[1], 0.0f);
    float T2 = fmaxf(temp[2], 0.0f);
    float c2 = (T0 + T2 - 2.0f * T1) * 0.25f;
    float coefs[3];
    coefs[0] = T0;              // X
    coefs[1] = (T1 - T0) + c2;  // LX
    coefs[2] = -c2;             // S2
    const float* mats[3] = { X, LX, S2 };

    // A layout (16x4 f32): lane -> M = lane%16, VGPR v holds K = v + 2*(lane/16)
    const size_t rowOff = (size_t)(R + l16) * 64 + Cb;

    v8f acc = {};
    #pragma unroll
    for (int m = 0; m < 3; ++m) {
        const float* V = mats[m];
        float cf = coefs[m];
        #pragma unroll
        for (int k = 0; k < 4; ++k) {
            int base = 4 * k + 2 * hi;
            v2f a = *(const v2f*)(V + rowOff + base);   // A[:,4k:4k+4] chunk
            // B layout (4x16 f32): VGPR v, half hi -> row j = v + 2*hi
            v2f b;
            b.x = (l16 == base)     ? cf : 0.0f;
            b.y = (l16 == base + 1) ? cf : 0.0f;
            acc = __builtin_amdgcn_wmma_f32_16x16x4_f32(
                false, a, false, b, (short)0, acc, false, false);
        }
    }

    // D layout (16x16 f32): VGPR v, lane -> row M = v + 8*hi, col N = l16
    #pragma unroll
    for (int v = 0; v < 8; ++v) {
        out[(size_t)(R + v + 8 * hi) * 64 + Cb + l16] = acc[v];
    }
}

// ---------------------------------------------------------------------------
// host orchestration
// ---------------------------------------------------------------------------
extern "C" void kernel_launch(void* const* d_in, const int* in_sizes, int n_in,
                              void* d_out, int out_size, void* d_ws, size_t ws_size,
                              hipStream_t stream) {
    const float* x    = (const float*)d_in[0];
    const int*   shuf = (const int*)  d_in[1];
    const int*   eidx = (const int*)  d_in[2];
    const float* ew   = (const float*)d_in[3];
    const int*   nidx = (const int*)  d_in[4];
    const float* nw   = (const float*)d_in[5];
    const float* temp = (const float*)d_in[6];

    const int N  = in_sizes[1];
    const int E  = in_sizes[3];
    const int En = in_sizes[5];
    const int D  = 64;
    const size_t ND = (size_t)N * D;

    float* out0 = (float*)d_out;        // out
    float* zpos = out0 + ND;            // z_pos
    float* zneg = out0 + 2 * ND;        // z_neg

    float* deg  = (float*)d_ws;                      // N floats (-> dis in place)
    size_t Na   = ((size_t)N + 63) & ~(size_t)63;
    float* bufA = deg + Na;                          // N*D scratch
    float* bufB = bufA + ND;                         // N*D scratch

    const int* erow = eidx;
    const int* ecol = eidx + E;
    const int* nrow = nidx;
    const int* ncol = nidx + En;

    dim3 blk(256);
    auto gridElems = [](size_t n) { return dim3((unsigned)((n + 255) / 256)); };
    dim3 gEdge((unsigned)((E  + 7) / 8));   // 8 waves/block, wave per edge
    dim3 gNb  ((unsigned)((En + 7) / 8));

    // 1) degree + D^{-1/2}
    zero_kernel   <<<gridElems(N),  blk, 0, stream>>>(deg, (size_t)N);
    deg_kernel    <<<gridElems(E),  blk, 0, stream>>>(erow, ew, deg, E);
    invsqrt_kernel<<<gridElems(N),  blk, 0, stream>>>(deg, N);

    // 2) S1 = Anorm @ x ;  Lx = x - S1  (bufB)
    zero_kernel     <<<gridElems(ND), blk, 0, stream>>>(bufA, ND);
    spmm_norm_kernel<<<gEdge,         blk, 0, stream>>>(erow, ecol, ew, deg, x, bufA, E);
    sub_kernel      <<<gridElems(ND), blk, 0, stream>>>(x, bufA, bufB, ND);

    // 3) S2 = Anorm @ Lx  (bufA)
    zero_kernel     <<<gridElems(ND), blk, 0, stream>>>(bufA, ND);
    spmm_norm_kernel<<<gEdge,         blk, 0, stream>>>(erow, ecol, ew, deg, bufB, bufA, E);

    // 4) out = T0*x + (T1-T0)*Lx + c2*(Lx - S2)  via WMMA
    int nTiles = (N / 16) * (D / 16);
    combine_wmma_kernel<<<dim3((unsigned)((nTiles + 7) / 8)), blk, 0, stream>>>(
        x, bufB, bufA, temp, out0, nTiles);

    // 5) z_pos = Nb @ (Nb @ out)
    zero_kernel   <<<gridElems(ND), blk, 0, stream>>>(bufA, ND);
    spmm_nb_kernel<<<gNb,           blk, 0, stream>>>(nrow, ncol, nw, nullptr, out0, bufA, En);
    zero_kernel   <<<gridElems(ND), blk, 0, stream>>>(zpos, ND);
    spmm_nb_kernel<<<gNb,           blk, 0, stream>>>(nrow, ncol, nw, nullptr, bufA, zpos, En);

    // 6) z_neg = Nb @ (Nb @ out[shuf])   (first hop gathers shuf[col])
    zero_kernel   <<<gridElems(ND), blk, 0, stream>>>(bufA, ND);
    spmm_nb_kernel<<<gNb,           blk, 0, stream>>>(nrow, ncol, nw, shuf, out0, bufA, En);
    zero_kernel   <<<gridElems(ND), blk, 0, stream>>>(zneg, ND);
    spmm_nb_kernel<<<gNb,           blk, 0, stream>>>(nrow, ncol, nw, nullptr, bufA, zneg, En);
}